// PrecomputedKDETimeEncoder_56959856280281
// MI455X (gfx1250) — compile-verified
//
#include <hip/hip_runtime.h>
#include <hip/hip_bf16.h>

// CDNA5 / gfx1250, wave32. fp32 WMMA path: V_WMMA_F32_16X16X4_F32,
// D(16x16,f32) = A(16x4,f32) x B(4x16,f32) + C.  K=20 = 5 k-tiles of 4.
//
// Roofline: 512 MB of output stores dominate (~22 us at 23.3 TB/s). The 5.12
// GFLOP projection rides the matrix pipe; cos fallback uses native v_cos_f32.
// Output + stream-once inputs use non-temporal hints so the hot 40 MB
// kde_table (and 10 KB W_proj) stay resident in the 192 MB L2.

typedef float v2f __attribute__((ext_vector_type(2)));
typedef float v4f __attribute__((ext_vector_type(4)));
typedef float v8f __attribute__((ext_vector_type(8)));
typedef int   v4i __attribute__((ext_vector_type(4)));

#define RKHS 20
#define OUTD 128

// ---------------------------------------------------------------------------
// Hot kernel: complete 16-row tiles only (numTiles = n/16). No bounds guards
// anywhere -> straight-line load / wmma / cos / select / store-clause body.
// ---------------------------------------------------------------------------
__global__ __launch_bounds__(256) void kde_full_kernel(
    const float* __restrict__ t_diff,     // [B]
    const int*   __restrict__ kde_idx,    // [B]
    const int*   __restrict__ kde_mask,   // [B] (0/1)
    const float* __restrict__ kde_table,  // [NUM_KDE, RKHS]
    const float* __restrict__ W_proj,     // [OUT, RKHS] row-major
    const float* __restrict__ b_proj,     // [OUT]
    const float* __restrict__ W_fb,       // [OUT, 1]
    const float* __restrict__ b_fb,       // [OUT]
    float*       __restrict__ out,        // [B, OUT]
    int numTiles)
{
    const int lane = threadIdx.x & 31;

    // Wave index is uniform but not provably so from threadIdx.x; pin it to an
    // SGPR so the tile bound test is a scalar branch (EXEC stays all-1s).
    const int wave = __builtin_amdgcn_readfirstlane((int)(threadIdx.x >> 5));
    const int tile = blockIdx.x * (blockDim.x >> 5) + wave;
    if (tile >= numTiles) return;        // scalar branch
    const int rowbase = tile * 16;

    const int m  = lane & 15;            // matrix row (A) / col (B,D) index
    const int hi = lane >> 4;            // lane half selects K pair / M half

    // ---- A fragments: gather 16 KDE rows. Lane layout (32-bit 16x4 A):
    //      VGPR0: K = 4kt + 2*hi,  VGPR1: K = 4kt + 2*hi + 1 -> one float2/lane/ktile.
    const long aidx = (long)__builtin_nontemporal_load(kde_idx + rowbase + m);
    const float* ap = kde_table + aidx * RKHS + hi * 2;
    v2f a[5];
#pragma unroll
    for (int kt = 0; kt < 5; ++kt)
        a[kt] = *(const v2f*)(ap + 4 * kt);

    // ---- fallback inputs for this lane's 8 result rows (D VGPR v <-> row rfb+v)
    const int rfb = rowbase + hi * 8;
    v4f td0 = __builtin_nontemporal_load((const v4f*)(t_diff + rfb));
    v4f td1 = __builtin_nontemporal_load((const v4f*)(t_diff + rfb + 4));
    v4i mk0 = __builtin_nontemporal_load((const v4i*)(kde_mask + rfb));
    v4i mk1 = __builtin_nontemporal_load((const v4i*)(kde_mask + rfb + 4));
    const float td[8] = { td0.x, td0.y, td0.z, td0.w, td1.x, td1.y, td1.z, td1.w };
    const int   mk[8] = { mk0.x, mk0.y, mk0.z, mk0.w, mk1.x, mk1.y, mk1.z, mk1.w };

    // Per-lane output base: row rfb, column offset added per col-tile.
    float* __restrict__ obase = out + (size_t)rfb * OUTD;

#pragma unroll
    for (int t = 0; t < OUTD / 16; ++t) {
        const int col = t * 16 + m;

        // C init = b_proj broadcast down the column
        const float bp = b_proj[col];
        v8f c;
#pragma unroll
        for (int i = 0; i < 8; ++i) c[i] = bp;

        // B fragments: B[k, n] = W_proj[col, k]; same K-pair-per-lane-half layout.
        const float* wp = W_proj + (long)col * RKHS + hi * 2;
#pragma unroll
        for (int kt = 0; kt < 5; ++kt) {
            v2f b = *(const v2f*)(wp + 4 * kt);
            // (neg_a, A, neg_b, B, c_mod, C, reuse_a, reuse_b)
            c = __builtin_amdgcn_wmma_f32_16x16x4_f32(
                    false, a[kt], false, b, (short)0, c, false, false);
        }

        // fused fallback + select + NT store clause (immediate offsets v*512B)
        const float wfb = W_fb[col];
        const float bfb = b_fb[col];
        float* __restrict__ op = obase + col;
#pragma unroll
        for (int v = 0; v < 8; ++v) {
            const float fb = __cosf(td[v] * wfb + bfb);   // native v_cos_f32
            const float r  = mk[v] ? c[v] : fb;
            __builtin_nontemporal_store(r, op + v * OUTD);
        }
    }
}

// ---------------------------------------------------------------------------
// Tail kernel: plain scalar path for the (n % 16) leftover rows. Launched only
// when a remainder exists (never for B = 1M).
// ---------------------------------------------------------------------------
__global__ __launch_bounds__(256) void kde_tail_kernel(
    const float* __restrict__ t_diff,
    const int*   __restrict__ kde_idx,
    const int*   __restrict__ kde_mask,
    const float* __restrict__ kde_table,
    const float* __restrict__ W_proj,
    const float* __restrict__ b_proj,
    const float* __restrict__ W_fb,
    const float* __restrict__ b_fb,
    float*       __restrict__ out,
    int rowstart, int n)
{
    const int idx   = blockIdx.x * blockDim.x + threadIdx.x;
    const int total = (n - rowstart) * OUTD;
    if (idx >= total) return;
    const int row = rowstart + idx / OUTD;
    const int col = idx % OUTD;

    float acc = b_proj[col];
    const long  ki = (long)kde_idx[row];
    const float* kv = kde_table + ki * RKHS;
    const float* wr = W_proj + (long)col * RKHS;
#pragma unroll
    for (int r = 0; r < RKHS; ++r)
        acc += kv[r] * wr[r];

    const float fb = __cosf(t_diff[row] * W_fb[col] + b_fb[col]);
    out[(size_t)row * OUTD + col] = kde_mask[row] ? acc : fb;
}

extern "C" void kernel_launch(void* const* d_in, const int* in_sizes, int n_in,
                              void* d_out, int out_size, void* d_ws, size_t ws_size,
                              hipStream_t stream) {
    // setup_inputs order: 0 src, 1 dst, 2 t_diff, 3 kde_idx, 4 kde_mask,
    //                     5 kde_table, 6 W_proj, 7 b_proj, 8 W_fb, 9 b_fb
    const float* t_diff    = (const float*)d_in[2];
    const int*   kde_idx   = (const int*)  d_in[3];
    const int*   kde_mask  = (const int*)  d_in[4];
    const float* kde_table = (const float*)d_in[5];
    const float* W_proj    = (const float*)d_in[6];
    const float* b_proj    = (const float*)d_in[7];
    const float* W_fb      = (const float*)d_in[8];
    const float* b_fb      = (const float*)d_in[9];
    float* out = (float*)d_out;

    const int n         = in_sizes[2];          // BATCH
    const int fullTiles = n / 16;               // complete 16-row tiles
    const int remRows   = n - fullTiles * 16;

    if (fullTiles > 0) {
        const int wavesPerBlock = 8;            // 256 threads = 8 wave32
        const int blocks = (fullTiles + wavesPerBlock - 1) / wavesPerBlock;
        kde_full_kernel<<<blocks, 256, 0, stream>>>(
            t_diff, kde_idx, kde_mask, kde_table, W_proj, b_proj, W_fb, b_fb,
            out, fullTiles);
    }
    if (remRows > 0) {                          // size-dependent only: capture-safe
        const int total  = remRows * OUTD;
        const int blocks = (total + 255) / 256;
        kde_tail_kernel<<<blocks, 256, 0, stream>>>(
            t_diff, kde_idx, kde_mask, kde_table, W_proj, b_proj, W_fb, b_fb,
            out, fullTiles * 16, n);
    }
}